// TAHE_28656021799675
// MI455X (gfx1250) — compile-verified
//
#include <hip/hip_runtime.h>
#include <hip/hip_bf16.h>

// TAHE user-history representation, MI455X (gfx1250), wave32.
// Memory-bound streaming kernel: ~846 MB traffic -> ~36us floor @ 23.3 TB/s.
// Phase 2 accumulation runs on V_WMMA_F32_16X16X4_F32 (the fp32 WMMA shape).

typedef __attribute__((ext_vector_type(2))) float v2f;
typedef __attribute__((ext_vector_type(8))) float v8f;

#define B_N 4096
#define L_N 200
#define D_N 128
#define EPS_L2 1e-12f

__global__ __launch_bounds__(256) void tahe_kernel(
    const float* __restrict__ rec,   // [B, L, D] recentTimeRepresentations
    const float* __restrict__ cur,   // [B, D]    curTimeRepresentation
    const int*   __restrict__ ts,    // [B, L]    recentTimestamps
    const float* __restrict__ emb,   // [B, L, D] recentItemEmbeddings
    float*       __restrict__ out)   // [B, D]
{
    __shared__ float s_curn[D_N];   // normalized current representation
    __shared__ float s_w[L_N];      // masked similarity weights
    __shared__ float s_part[8];     // cross-wave reduction scratch

    const int b    = blockIdx.x;
    const int tid  = threadIdx.x;
    const int wave = tid >> 5;
    const int lane = tid & 31;

    const float* __restrict__ curb = cur + (size_t)b * D_N;
    const float* __restrict__ recb = rec + (size_t)b * L_N * D_N;
    const float* __restrict__ embb = emb + (size_t)b * L_N * D_N;
    const int*   __restrict__ tsb  = ts  + (size_t)b * L_N;

    // ---------- Phase 0: cur_n = l2_normalize(cur[b]) ----------
    // Threads 0..127 (waves 0..3) each own one d.
    if (tid < D_N) {
        float x  = curb[tid];
        float xx = x * x;
        #pragma unroll
        for (int off = 16; off > 0; off >>= 1)
            xx += __shfl_xor(xx, off, 32);
        if (lane == 0) s_part[wave] = xx;
    }
    __syncthreads();
    {
        float ssq = s_part[0] + s_part[1] + s_part[2] + s_part[3];
        float rs  = __frsqrt_rn(fmaxf(ssq, EPS_L2));
        if (tid < D_N) s_curn[tid] = curb[tid] * rs;
    }
    __syncthreads();

    // Per-lane chunk of cur_n for phase 1 (lane owns d = lane*4 .. lane*4+3)
    float4 cn = *(const float4*)(&s_curn[lane * 4]);

    // ---------- Phase 1: w[l] = mask * (cos_sim + 1) * 0.5 ----------
    // Wave w handles l = w, w+8, ... (25 rows each). A wave reads one
    // contiguous 512B row chunk per iteration (float4/lane, fully coalesced).
    for (int l = wave; l < L_N; l += 8) {
        float4 r = *(const float4*)(recb + (size_t)l * D_N + lane * 4);
        float ss = r.x * r.x + r.y * r.y + r.z * r.z + r.w * r.w;
        float dc = r.x * cn.x + r.y * cn.y + r.z * cn.z + r.w * cn.w;
        #pragma unroll
        for (int off = 16; off > 0; off >>= 1) {
            ss += __shfl_xor(ss, off, 32);
            dc += __shfl_xor(dc, off, 32);
        }
        if (lane == 0) {
            float sim = dc * __frsqrt_rn(fmaxf(ss, EPS_L2));
            float m   = (tsb[l] > 0) ? 1.0f : 0.0f;
            s_w[l]    = m * (sim + 1.0f) * 0.5f;
        }
    }
    __syncthreads();

    // ---------- Phase 2: out[d] = sum_l w[l] * E[l,d] via WMMA fp32 ----------
    // Wave w owns d tile [w*16, w*16+16). C = A(16x4) x B(4x16) + C, where
    // A row m == (w[k0], w[k0+1], w[k0+2], w[k0+3]) for every m (broadcast),
    // so every row of C equals the output tile; we emit row M=0.
    //
    // A layout (ISA 7.12.2, 32-bit A 16x4): lanes 0-15 hold K=0 (VGPR0) and
    // K=1 (VGPR1); lanes 16-31 hold K=2,3.  B 4x16 mirrors it row-striped:
    // VGPR0 = rows K=0 (lanes 0-15) / K=2 (lanes 16-31), VGPR1 = K=1 / K=3.
    {
        const int d0   = wave * 16;
        const int half = lane >> 4;       // 0: K=0,1   1: K=2,3
        const int n    = lane & 15;       // column within tile

        v8f c = {};
        for (int k0 = 0; k0 < L_N; k0 += 4) {
            v2f a, bm;
            a.x = s_w[k0 + 2 * half + 0];     // LDS broadcast reads
            a.y = s_w[k0 + 2 * half + 1];
            const float* p = embb + (size_t)(k0 + 2 * half) * D_N + d0 + n;
            bm.x = p[0];                      // row k0+2*half
            bm.y = p[D_N];                    // row k0+2*half+1
            // (neg_a, A, neg_b, B, c_mod, C, reuse_a, reuse_b)
            c = __builtin_amdgcn_wmma_f32_16x16x4_f32(
                    false, a, false, bm, (short)0, c, false, false);
        }
        if (lane < 16)
            out[(size_t)b * D_N + d0 + lane] = c[0];   // M=0 row
    }
}

extern "C" void kernel_launch(void* const* d_in, const int* in_sizes, int n_in,
                              void* d_out, int out_size, void* d_ws, size_t ws_size,
                              hipStream_t stream) {
    (void)in_sizes; (void)n_in; (void)out_size; (void)d_ws; (void)ws_size;
    const float* rec = (const float*)d_in[0];  // recentTimeRepresentations [B,L,D]
    const float* cur = (const float*)d_in[1];  // curTimeRepresentation    [B,D]
    const int*   ts  = (const int*)  d_in[2];  // recentTimestamps         [B,L]
    const float* emb = (const float*)d_in[3];  // recentItemEmbeddings     [B,L,D]
    float*       out = (float*)d_out;          // [B,D]

    tahe_kernel<<<B_N, 256, 0, stream>>>(rec, cur, ts, emb, out);
}